// DistributionalCritic_50826642981360
// MI455X (gfx1250) — compile-verified
//
#include <hip/hip_runtime.h>
#include <stdint.h>

// ---------------------------------------------------------------------------
// C51 distributional projection for MI455X (gfx1250).
// Memory-bound one-pass stream (~218 MB -> ~9.4 us floor @ 23.3 TB/s).
//   * async global->LDS staging of a 128x51 f32 tile (coalesced B128, NT hint)
//   * each thread owns one row: row -> 51 VGPRs, zero row in LDS, then dual
//     scatter-add back into the SAME tile row via ds_add_f32 LDS atomics
//     (stride 51, gcd(51,64)=1 -> conflict-free for the fixed-offset phases)
//   * async LDS->global B128 stores of the finished tile (NT hint)
// Single 26 KB LDS buffer -> occupancy at the 32-wave/WGP cap.
// ---------------------------------------------------------------------------

namespace {
constexpr int   kAtoms        = 51;
constexpr int   kRowsPerBlock = 128;   // 4 waves (wave32)
constexpr float kVMin         = -10.0f;
constexpr float kVMax         =  10.0f;
constexpr float kDiscount     =  0.99f;
constexpr float kDeltaZ       = (kVMax - kVMin) / float(kAtoms - 1);
constexpr float kInvDz        = 1.0f / kDeltaZ;
}

// Low 32 bits of a generic pointer into LDS == workgroup LDS byte offset
// (flat-aperture layout, ISA 10.2).
static __device__ __forceinline__ unsigned lds_byte_off(const void* p) {
  return (unsigned)(uintptr_t)p;
}

static __device__ __forceinline__ void async_load_b128(unsigned ldsOff, const void* g) {
  asm volatile("global_load_async_to_lds_b128 %0, %1, off th:TH_LOAD_NT"
               :: "v"(ldsOff), "v"((unsigned long long)(uintptr_t)g)
               : "memory");
}
static __device__ __forceinline__ void async_load_b32(unsigned ldsOff, const void* g) {
  asm volatile("global_load_async_to_lds_b32 %0, %1, off th:TH_LOAD_NT"
               :: "v"(ldsOff), "v"((unsigned long long)(uintptr_t)g)
               : "memory");
}
static __device__ __forceinline__ void async_store_b128(void* g, unsigned ldsOff) {
  asm volatile("global_store_async_from_lds_b128 %0, %1, off th:TH_STORE_NT"
               :: "v"((unsigned long long)(uintptr_t)g), "v"(ldsOff)
               : "memory");
}
static __device__ __forceinline__ void async_store_b32(void* g, unsigned ldsOff) {
  asm volatile("global_store_async_from_lds_b32 %0, %1, off th:TH_STORE_NT"
               :: "v"((unsigned long long)(uintptr_t)g), "v"(ldsOff)
               : "memory");
}
static __device__ __forceinline__ void wait_async0() {
  asm volatile("s_wait_asynccnt 0x0" ::: "memory");
}

__global__ __launch_bounds__(kRowsPerBlock)
void c51_project_kernel(const float* __restrict__ next_dist,
                        const float* __restrict__ rewards,
                        const float* __restrict__ bootstrap,
                        float* __restrict__ out,
                        int rows)
{
  // Single staging + in-place accumulation tile: 128*51*4 = 26112 B
  __shared__ __align__(16) float s_tile[kRowsPerBlock * kAtoms];

  const int tid   = threadIdx.x;
  const int row0  = blockIdx.x * kRowsPerBlock;
  int rowsInBlock = rows - row0;
  if (rowsInBlock <= 0) return;                 // uniform per block
  if (rowsInBlock > kRowsPerBlock) rowsInBlock = kRowsPerBlock;

  const int tileFloats = rowsInBlock * kAtoms;  // 6528 for a full block
  const int nChunk4    = tileFloats >> 2;       // 16B chunks; tile base offset
                                                // 128*51*4 = 26112 ≡ 0 mod 16
  const unsigned tileOff = lds_byte_off(s_tile);
  const float* gsrc = next_dist + (size_t)row0 * kAtoms;

  // ---- stage input tile: coalesced async global->LDS (non-temporal) ----
  for (int e = tid; e < nChunk4; e += kRowsPerBlock)
    async_load_b128(tileOff + (unsigned)e * 16u, gsrc + (size_t)e * 4);
  for (int e = (nChunk4 << 2) + tid; e < tileFloats; e += kRowsPerBlock)
    async_load_b32(tileOff + (unsigned)e * 4u, gsrc + e);

  wait_async0();
  __syncthreads();

  // ---- per-row projection, accumulating in place in the tile row ----
  if (tid < rowsInBlock) {
    const int   row = row0 + tid;
    const float r   = __builtin_nontemporal_load(rewards + row);
    const float bg  = __builtin_nontemporal_load(bootstrap + row) * kDiscount;

    float* a = &s_tile[tid * kAtoms];

    // Pull the row into registers (static indices -> VGPRs), zero the row.
    float nd[kAtoms];
#pragma unroll
    for (int k = 0; k < kAtoms; ++k) {
      nd[k] = a[k];
      a[k]  = 0.0f;
    }
    // Same-thread DS ops are in-order (DScnt pipe), so the zeroing above is
    // complete w.r.t. the atomics below.
#pragma unroll
    for (int j = 0; j < kAtoms; ++j) {
      const float zj = kVMin + kDeltaZ * (float)j;
      float tz = r + bg * zj;
      tz = fminf(fmaxf(tz, kVMin), kVMax);
      const float b = (tz - kVMin) * kInvDz;         // in [0, 50]
      int lo = (int)floorf(b);
      int hi = (int)ceilf(b);
      if (lo == hi) {                                 // exact-integer bin
        if (lo > 0) lo -= 1; else hi += 1;
      }
      const float p = nd[j];
      atomicAdd(&a[lo], p * ((float)hi - b));         // ds_add_f32 (no-rtn)
      atomicAdd(&a[hi], p * (b - (float)lo));
    }
  }

  __syncthreads();

  // ---- coalesced async LDS->global store of the finished tile (NT) ----
  float* gdst = out + (size_t)row0 * kAtoms;
  for (int e = tid; e < nChunk4; e += kRowsPerBlock)
    async_store_b128(gdst + (size_t)e * 4, tileOff + (unsigned)e * 16u);
  for (int e = (nChunk4 << 2) + tid; e < tileFloats; e += kRowsPerBlock)
    async_store_b32(gdst + e, tileOff + (unsigned)e * 4u);

  wait_async0();   // S_ENDPGM implies wait-idle; explicit for clarity
}

extern "C" void kernel_launch(void* const* d_in, const int* in_sizes, int n_in,
                              void* d_out, int out_size, void* d_ws, size_t ws_size,
                              hipStream_t stream) {
  (void)n_in; (void)out_size; (void)d_ws; (void)ws_size;
  const float* next_dist = (const float*)d_in[0];
  const float* rewards   = (const float*)d_in[1];
  const float* bootstrap = (const float*)d_in[2];
  float*       out       = (float*)d_out;

  const int rows   = in_sizes[1];                       // B (rewards is [B])
  const int blocks = (rows + kRowsPerBlock - 1) / kRowsPerBlock;

  c51_project_kernel<<<dim3(blocks), dim3(kRowsPerBlock), 0, stream>>>(
      next_dist, rewards, bootstrap, out, rows);
}